// LiDAR_Renderer_22978075034091
// MI455X (gfx1250) — compile-verified
//
#include <hip/hip_runtime.h>
#include <math.h>

typedef __attribute__((ext_vector_type(2))) float v2f;
typedef __attribute__((ext_vector_type(8))) float v8f;

#define NEAR_LIDAR 0.01f
#define SPAN       0.80f
#define BOUND      1.0f
#define WAVES      4

// D = A(16x4,f32) * B(4x16,f32) + C(16x16,f32)  — CDNA5 V_WMMA_F32_16X16X4_F32
__device__ __forceinline__ v8f wmma4(v2f a, v2f b, v8f c) {
  return __builtin_amdgcn_wmma_f32_16x16x4_f32(
      /*neg_a=*/false, a, /*neg_b=*/false, b,
      /*c_mod=*/(short)0, c, /*reuse_a=*/false, /*reuse_b=*/false);
}

// single-instruction ReLU: avoids fmaxf's max(x,x) sNaN-canonicalization op
__device__ __forceinline__ float reluf(float x) {
  float r;
  asm("v_max_num_f32 %0, %1, 0" : "=v"(r) : "v"(x));
  return r;
}

// wave-private LDS staging fence: DS ops are in-order per wave; the asm +
// memory clobber stops the compiler from reordering around the staging.
__device__ __forceinline__ void lds_fence() {
  __builtin_amdgcn_wave_barrier();
  asm volatile("s_wait_dscnt 0" ::: "memory");
  __builtin_amdgcn_wave_barrier();
}

// branchless, numerically stable softplus: keeps EXEC all-ones (WMMA req.)
__device__ __forceinline__ float softplusf(float x) {
  return fmaxf(x, 0.f) + __logf(1.f + __expf(-fabsf(x)));
}
__device__ __forceinline__ float sigmoidf(float x) {
  return 1.f / (1.f + __expf(-x));
}

__global__ __launch_bounds__(WAVES * 32)
void lidar_render_wmma_kernel(
    const float* __restrict__ rays_o, const float* __restrict__ rays_d,
    const float* __restrict__ timep,  const float* __restrict__ W1,
    const float* __restrict__ b1,     const float* __restrict__ Wsig,
    const float* __restrict__ Wfeat,  const float* __restrict__ Wa1,
    const float* __restrict__ ba1,    const float* __restrict__ Wa2,
    const int*   __restrict__ nstepp, float* __restrict__ out, int N)
{
  __shared__ float hbuf [WAVES][16][32];  // h rows, row-major  (A-frag reload)
  __shared__ float xabuf[WAVES][16][24];  // [xyz dir feat pad2]
  __shared__ float gbuf [WAVES][16][32];  // attr hidden rows

  const int tid  = threadIdx.x;
  const int w    = tid >> 5;
  const int lane = tid & 31;
  const int nl   = lane & 15;     // N-index / M-index inside tile
  const int hi   = lane >> 4;     // half-wave: K split for A/B frags
  const int k0   = hi * 2;        // frag rows: VGPR0->K=k0, VGPR1->K=k0+1
  const int rayBase = blockIdx.x * (WAVES * 16) + w * 16;
  int ray = rayBase + nl; if (ray >= N) ray = N - 1;   // uniform clamp

  const int   S      = *nstepp;
  const float tval   = timep[0];
  const float invSm1 = 1.f / (float)(S - 1);
  const float dlt    = SPAN * invSm1;        // deltas[0..S-2]
  const float dltL   = SPAN / (float)S;      // sample_dist (last delta)

  const float rox = rays_o[ray*3+0], roy = rays_o[ray*3+1], roz = rays_o[ray*3+2];
  const float rdx = rays_d[ray*3+0], rdy = rays_d[ray*3+1], rdz = rays_d[ray*3+2];

  // ---------- preload B fragments ----------
  v2f B1t0, B1t1;                               // W1 [4][32], two N tiles
  B1t0.x = W1[(k0+0)*32 + nl];     B1t0.y = W1[(k0+1)*32 + nl];
  B1t1.x = W1[(k0+0)*32 + nl+16];  B1t1.y = W1[(k0+1)*32 + nl+16];

  v2f wfB[8];                                   // Wfeat [32][16], K in 8 chunks
  #pragma unroll
  for (int c = 0; c < 8; ++c) {
    wfB[c].x = Wfeat[(4*c+k0+0)*16 + nl];
    wfB[c].y = Wfeat[(4*c+k0+1)*16 + nl];
  }

  v2f wsB[8];                                   // Wsig as B: cols 0 AND 1 = Wsig
  #pragma unroll
  for (int c = 0; c < 8; ++c) {
    const float s0 = Wsig[4*c + k0 + 0];
    const float s1 = Wsig[4*c + k0 + 1];
    wsB[c].x = (nl <= 1) ? s0 : 0.f;
    wsB[c].y = (nl <= 1) ? s1 : 0.f;
  }

  v2f wa2B[8];                                  // Wa2 [32][2] as B: col0,col1, rest 0
  #pragma unroll
  for (int c = 0; c < 8; ++c) {
    const int j0 = 4*c + k0, j1 = j0 + 1;
    wa2B[c].x = (nl <= 1) ? Wa2[j0*2 + nl] : 0.f;
    wa2B[c].y = (nl <= 1) ? Wa2[j1*2 + nl] : 0.f;
  }

  v2f wa1B[6][2];                               // Wa1 [22][32] padded to K=24
  #pragma unroll
  for (int c = 0; c < 6; ++c)
    #pragma unroll
    for (int t = 0; t < 2; ++t) {
      const int r0 = 4*c + k0, r1 = r0 + 1;
      wa1B[c][t].x = (r0 < 22) ? Wa1[r0*32 + nl + 16*t] : 0.f;
      wa1B[c][t].y = (r1 < 22) ? Wa1[r1*32 + nl + 16*t] : 0.f;
    }

  // loop-invariant bias C tiles (materialized once in the preheader)
  v8f cb0, cb1, cba0, cba1;
  {
    const float b1v0  = b1[nl],  b1v1  = b1[nl+16];
    const float ba1v0 = ba1[nl], ba1v1 = ba1[nl+16];
    #pragma unroll
    for (int r = 0; r < 8; ++r) {
      cb0[r] = b1v0;  cb1[r] = b1v1;
      cba0[r] = ba1v0; cba1[r] = ba1v1;
    }
  }

  xabuf[w][nl][22] = 0.f;                       // K-padding rows (written once)
  xabuf[w][nl][23] = 0.f;

  // per-ray scan state: meaningful on lanes n<=1 (n=0: depth/wsum/img0, n=1: img1)
  float T[8], wsum[8], dep[8], imv[8];
  #pragma unroll
  for (int r = 0; r < 8; ++r) { T[r]=1.f; wsum[r]=0.f; dep[r]=0.f; imv[r]=0.f; }

  for (int s = 0; s < S; ++s) {
    const float zv = NEAR_LIDAR + SPAN * (float)s * invSm1;
    const float px = fminf(fmaxf(rox + rdx*zv, -BOUND), BOUND);
    const float py = fminf(fmaxf(roy + rdy*zv, -BOUND), BOUND);
    const float pz = fminf(fmaxf(roz + rdz*zv, -BOUND), BOUND);

    // ---- layer 1: [16 rays x 4(xyzt)] @ W1 -> h (two 16x16 D tiles), +bias, relu
    v2f A1; A1.x = hi ? pz : px; A1.y = hi ? tval : py;
    v8f h0 = wmma4(A1, B1t0, cb0);
    v8f h1 = wmma4(A1, B1t1, cb1);
    #pragma unroll
    for (int r = 0; r < 8; ++r) { h0[r] = reluf(h0[r]); h1[r] = reluf(h1[r]); }

    // ---- stage h: D layout (lane=N, vgpr=M) -> LDS row-major [m][32]
    #pragma unroll
    for (int r = 0; r < 8; ++r) {
      const int m = r + 8*hi;
      hbuf[w][m][nl]    = h0[r];
      hbuf[w][m][nl+16] = h1[r];
    }
    lds_fence();

    // ---- feat = h @ Wfeat  AND  sigma-dot = h @ [Wsig,Wsig,0..]
    //      two independent 8-WMMA chains sharing the same A fragments
    v8f ft, fs;
    #pragma unroll
    for (int r = 0; r < 8; ++r) { ft[r] = 0.f; fs[r] = 0.f; }
    #pragma unroll
    for (int c = 0; c < 8; ++c) {
      v2f A; A.x = hbuf[w][nl][4*c + k0]; A.y = hbuf[w][nl][4*c + k0 + 1];
      ft = wmma4(A, wfB[c], ft);
      fs = wmma4(A, wsB[c], fs);
    }

    // ---- sigma/alpha/transmittance scan (valid on lanes n<=1; same value both)
    float wgt[8];
    const float dcur = (s == S-1) ? dltL : dlt;
    #pragma unroll
    for (int r = 0; r < 8; ++r) {
      const float sg = softplusf(fs[r]);
      const float al = 1.f - __expf(-dcur * sg);
      const float wv = al * T[r];
      T[r]    *= (1.f - al + 1e-15f);
      wgt[r]   = wv;
      wsum[r] += wv;
      dep[r]  += wv * zv;
    }

    // ---- stage attr input xa = [xyz(3) dir(3) feat(16) pad(2)]
    xabuf[w][nl][0] = px;  xabuf[w][nl][1] = py;  xabuf[w][nl][2] = pz;
    xabuf[w][nl][3] = rdx; xabuf[w][nl][4] = rdy; xabuf[w][nl][5] = rdz;
    #pragma unroll
    for (int r = 0; r < 8; ++r) {
      const int m = r + 8*hi;
      xabuf[w][m][6 + nl] = ft[r];
    }
    lds_fence();

    // ---- attr layer 1: [16 x 24] @ Wa1(24x32), two independent D-tile chains
    v8f g0 = cba0, g1 = cba1;
    #pragma unroll
    for (int c = 0; c < 6; ++c) {
      v2f A; A.x = xabuf[w][nl][4*c + k0]; A.y = xabuf[w][nl][4*c + k0 + 1];
      g0 = wmma4(A, wa1B[c][0], g0);
      g1 = wmma4(A, wa1B[c][1], g1);
    }
    #pragma unroll
    for (int r = 0; r < 8; ++r) { g0[r] = reluf(g0[r]); g1[r] = reluf(g1[r]); }

    // ---- stage g, then attr layer 2 as one 8-WMMA chain:
    //      q = g @ [Wa2col0, Wa2col1, 0..] ; out0 at lane n=0, out1 at lane n=1
    #pragma unroll
    for (int r = 0; r < 8; ++r) {
      const int m = r + 8*hi;
      gbuf[w][m][nl]    = g0[r];
      gbuf[w][m][nl+16] = g1[r];
    }
    lds_fence();

    v8f qd;
    #pragma unroll
    for (int r = 0; r < 8; ++r) qd[r] = 0.f;
    #pragma unroll
    for (int c = 0; c < 8; ++c) {
      v2f A; A.x = gbuf[w][nl][4*c + k0]; A.y = gbuf[w][nl][4*c + k0 + 1];
      qd = wmma4(A, wa2B[c], qd);
    }

    // ---- sigmoid + weighted accumulation (lanes n=0 -> img0, n=1 -> img1)
    #pragma unroll
    for (int r = 0; r < 8; ++r)
      imv[r] += wgt[r] * sigmoidf(qd[r]);
  }

  // ---- outputs: depth[N] | image[N][2] | weights_sum[N]
  #pragma unroll
  for (int r = 0; r < 8; ++r) {
    const int m  = r + 8*hi;
    const int rr = rayBase + m;
    if (rr < N) {
      if (nl == 0) {
        out[rr]           = dep[r];
        out[N + rr*2 + 0] = imv[r];
        out[3*N + rr]     = wsum[r];
      } else if (nl == 1) {
        out[N + rr*2 + 1] = imv[r];
      }
    }
  }
}

extern "C" void kernel_launch(void* const* d_in, const int* in_sizes, int n_in,
                              void* d_out, int out_size, void* d_ws, size_t ws_size,
                              hipStream_t stream) {
  const float* rays_o = (const float*)d_in[0];
  const float* rays_d = (const float*)d_in[1];
  const float* timep  = (const float*)d_in[2];
  const float* W1     = (const float*)d_in[3];
  const float* b1     = (const float*)d_in[4];
  const float* Wsig   = (const float*)d_in[5];
  const float* Wfeat  = (const float*)d_in[6];
  const float* Wa1    = (const float*)d_in[7];
  const float* ba1    = (const float*)d_in[8];
  const float* Wa2    = (const float*)d_in[9];
  const int*   nstep  = (const int*)d_in[10];

  const int N = in_sizes[0] / 3;                 // 8192 rays
  const int raysPerBlock = WAVES * 16;           // 64
  const int blocks = (N + raysPerBlock - 1) / raysPerBlock;

  lidar_render_wmma_kernel<<<blocks, WAVES * 32, 0, stream>>>(
      rays_o, rays_d, timep, W1, b1, Wsig, Wfeat, Wa1, ba1, Wa2,
      nstep, (float*)d_out, N);
}